// WARPLoss_76630806495450
// MI455X (gfx1250) — compile-verified
//
#include <hip/hip_runtime.h>

typedef __attribute__((ext_vector_type(2))) float v2f;
typedef __attribute__((ext_vector_type(8))) float v8f;

// ---------------------------------------------------------------------------
// Stage 1: one wave32 per row. Lane l handles trials l and l+32 (T = 64 fast
// path, guard-free). ballot_w32 -> first accepting trial; readlane (uniform
// lane id) broadcasts its score. Rows [B, Bpad) write 0 so stage 2 can run
// guard-free over a multiple of 64.
// ---------------------------------------------------------------------------
__global__ void warp_row_kernel(const float* __restrict__ inp,
                                const int* __restrict__ pos_idx,
                                const int* __restrict__ neg_cands,
                                float* __restrict__ row_loss,
                                int B, int Bpad, int Y, int T) {
    const int gwave = (int)((blockIdx.x * blockDim.x + threadIdx.x) >> 5);
    const int lane  = (int)(threadIdx.x & 31);
    if (gwave >= Bpad) return;              // wave-uniform
    if (gwave >= B) {                       // zero-pad for the WMMA reduction
        if (lane == 0) row_loss[gwave] = 0.0f;
        return;
    }

    const size_t rbase = (size_t)gwave * (size_t)Y;
    const float  pos   = inp[rbase + (size_t)pos_idx[gwave]];
    const float  thr   = pos - 1.0f;        // accept iff neg >= pos - 1

    int   first = -1;
    float nsel  = 0.0f;

    if (T == 64) {
        // Fast path: no guards, all gathers issued before any dependency.
        const size_t cbase = (size_t)gwave * 64u;
        const int   c0 = neg_cands[cbase + lane];
        const int   c1 = neg_cands[cbase + lane + 32];
        const float n0 = inp[rbase + (size_t)c0];
        const float n1 = inp[rbase + (size_t)c1];

        const unsigned m0 = __builtin_amdgcn_ballot_w32(n0 >= thr);
        const unsigned m1 = __builtin_amdgcn_ballot_w32(n1 >= thr);
        if (m0) {                           // uniform: ballot result is scalar
            const int l = __builtin_ctz(m0);
            first = l;
            nsel  = __int_as_float(__builtin_amdgcn_readlane(__float_as_int(n0), l));
        } else if (m1) {
            const int l = __builtin_ctz(m1);
            first = 32 + l;
            nsel  = __int_as_float(__builtin_amdgcn_readlane(__float_as_int(n1), l));
        }
    } else {
        for (int tb = 0; tb < T; tb += 64) {
            const int t0 = tb + lane;
            const int t1 = tb + lane + 32;
            const size_t cbase = (size_t)gwave * (size_t)T;
            const int   c0 = (t0 < T) ? neg_cands[cbase + t0] : 0;
            const int   c1 = (t1 < T) ? neg_cands[cbase + t1] : 0;
            const float n0 = (t0 < T) ? inp[rbase + (size_t)c0] : -3.4e38f;
            const float n1 = (t1 < T) ? inp[rbase + (size_t)c1] : -3.4e38f;

            const unsigned m0 = __builtin_amdgcn_ballot_w32(n0 >= thr);
            const unsigned m1 = __builtin_amdgcn_ballot_w32(n1 >= thr);
            if (m0) {
                const int l = __builtin_ctz(m0);
                first = tb + l;
                nsel  = __int_as_float(__builtin_amdgcn_readlane(__float_as_int(n0), l));
                break;
            }
            if (m1) {
                const int l = __builtin_ctz(m1);
                first = tb + 32 + l;
                nsel  = __int_as_float(__builtin_amdgcn_readlane(__float_as_int(n1), l));
                break;
            }
        }
    }

    float loss = 0.0f;
    if (first >= 0) {
        const float num_trials = (float)(first + 1);
        const float L = logf(floorf(((float)Y - 1.0f) / num_trials));
        loss = L * (1.0f - pos + nsel);
    }
    if (lane == 0) row_loss[gwave] = loss;
}

// ---------------------------------------------------------------------------
// Stage 2: deterministic f32 sum of npad (multiple of 64) values with
// V_WMMA_F32_16X16X4_F32. A = ones(16x4) -> D[m][n] = sum_k B[k][n] + C[m][n];
// each WMMA folds 64 values (one global_load_b64 per lane) into the f32
// accumulator. Row 0 of D (c[0], lanes 0..15) holds 16 column sums; a 16-lane
// xor-fold finishes. Single wave, EXEC all-ones, uniform control flow.
// ---------------------------------------------------------------------------
__global__ void reduce_wmma_kernel(const float* __restrict__ vals,
                                   float* __restrict__ out, int npad) {
    const int lane = (int)(threadIdx.x & 31);
    v2f a; a.x = 1.0f; a.y = 1.0f;          // all-ones A matrix (16x4 f32)
    v8f c = {};                              // f32 accumulator (16x16)

    const v2f* __restrict__ p = (const v2f*)vals + lane;  // 2 floats per lane
    const int iters = npad >> 6;             // 64 values per WMMA
    for (int i = 0; i < iters; ++i) {        // uniform scalar loop
        const v2f b = p[i * 32];             // guard-free global_load_b64
        c = __builtin_amdgcn_wmma_f32_16x16x4_f32(
                false, a, false, b, (short)0, c, false, false);
    }

    float s = c[0];
    s += __shfl_xor(s, 1, 32);
    s += __shfl_xor(s, 2, 32);
    s += __shfl_xor(s, 4, 32);
    s += __shfl_xor(s, 8, 32);              // lane 0 = sum of lanes 0..15
    if (lane == 0) out[0] = s;
}

// ---------------------------------------------------------------------------
// Launch: inputs are (input[B,Y] f32, target[B,Y] i32 (unused), pos_idx[B] i32,
// neg_cands[B,T] i32). Output: one f32 scalar.
// ---------------------------------------------------------------------------
extern "C" void kernel_launch(void* const* d_in, const int* in_sizes, int n_in,
                              void* d_out, int out_size, void* d_ws, size_t ws_size,
                              hipStream_t stream) {
    const float* inp     = (const float*)d_in[0];
    const int*   pos_idx = (const int*)d_in[2];
    const int*   neg     = (const int*)d_in[3];
    const int B    = in_sizes[2];
    const int Y    = in_sizes[0] / B;
    const int T    = in_sizes[3] / B;
    const int Bpad = (B + 63) & ~63;        // multiple of 64 for stage 2

    float* row_loss = (float*)d_ws;         // Bpad floats of scratch

    const int waves_per_block = 8;          // 256 threads = 8 wave32
    const int threads = waves_per_block * 32;
    const int blocks  = (Bpad + waves_per_block - 1) / waves_per_block;

    warp_row_kernel<<<blocks, threads, 0, stream>>>(inp, pos_idx, neg,
                                                    row_loss, B, Bpad, Y, T);
    reduce_wmma_kernel<<<1, 32, 0, stream>>>(row_loss, (float*)d_out, Bpad);
}